// Z1SDPSlover_43868795961577
// MI455X (gfx1250) — compile-verified
//
#include <hip/hip_runtime.h>
#include <hip/hip_bf16.h>
#include <math.h>

// ---------------- problem constants ----------------
#define Nn  70
#define Mm  85
#define NT  2485
#define NP  80      // N padded to multiple of 16
#define MP  96      // M padded
#define NTP 2496    // NT padded

#define DSQ2 1.4142135623730951
#define DSQH 0.7071067811865476

// scalar slots (device-resident control values; no host readback -> graph safe)
#define S_KSI 0
#define S_ETA 1
#define S_NORMSA 2
#define S_NORMB 3
#define S_NORMC 4
#define S_MU 5
#define S_XZ 6
#define S_LAM 7
#define S_ALPHA 8
#define S_BETA 9
#define S_MAB 10
#define S_SIGMA 11
#define S_G2 12
#define S_INNER 13

typedef float v2f __attribute__((ext_vector_type(2)));
typedef float v8f __attribute__((ext_vector_type(8)));

// ---------------- device helpers ----------------
__device__ __forceinline__ void tril_rc(int k, int& r, int& c) {
  int rr = (int)((sqrtf(8.0f * (float)k + 1.0f) - 1.0f) * 0.5f);
  while ((rr + 1) * (rr + 2) / 2 <= k) ++rr;
  while (rr * (rr + 1) / 2 > k) --rr;
  r = rr; c = k - rr * (rr + 1) / 2;
}

__device__ __forceinline__ double blockSum256(double v) {
  __shared__ double red[256];
  int t = threadIdx.x;
  red[t] = v; __syncthreads();
  for (int s = 128; s > 0; s >>= 1) { if (t < s) red[t] += red[t + s]; __syncthreads(); }
  double r = red[0]; __syncthreads();
  return r;
}

__device__ __forceinline__ double blockMax256(double v) {
  __shared__ double redm[256];
  int t = threadIdx.x;
  redm[t] = v; __syncthreads();
  for (int s = 128; s > 0; s >>= 1) { if (t < s) redm[t] = fmax(redm[t], redm[t + s]); __syncthreads(); }
  double r = redm[0]; __syncthreads();
  return r;
}

// ---------------- WMMA fp32 GEMM (16x16 tile / wave32, K step 4) ----------------
// C[l] (mp x np) = A[l] (mp x kp) @ B[l] (kp x np); strides allow operand broadcast.
__global__ void wmma_bgemm_k(const float* __restrict__ A, const float* __restrict__ B,
                             float* __restrict__ C, int mp, int kp, int np,
                             long sA, long sB, long sC) {
  int l = blockIdx.z;
  const float* Ab = A + (long)l * sA;
  const float* Bb = B + (long)l * sB;
  float* Cb = C + (long)l * sC;
  int row0 = blockIdx.x * 16, col0 = blockIdx.y * 16;
  int lane = threadIdx.x & 31;
  int half = lane >> 4;        // 0: K=0,1   1: K=2,3   (ISA 7.12.2, 32-bit A 16x4)
  int lr   = lane & 15;
  v8f acc = {0.f, 0.f, 0.f, 0.f, 0.f, 0.f, 0.f, 0.f};
  for (int k0 = 0; k0 < kp; k0 += 4) {
    int ka = k0 + 2 * half;
    v2f a, b;
    a.x = Ab[(long)(row0 + lr) * kp + ka];
    a.y = Ab[(long)(row0 + lr) * kp + ka + 1];
    b.x = Bb[(long)ka * np + col0 + lr];
    b.y = Bb[(long)(ka + 1) * np + col0 + lr];
    acc = __builtin_amdgcn_wmma_f32_16x16x4_f32(false, a, false, b, (short)0, acc,
                                                false, false);
  }
#pragma unroll
  for (int v = 0; v < 8; ++v)   // C/D: VGPR v -> row v (lanes 0-15) / row v+8 (16-31)
    Cb[(long)(row0 + v + 8 * half) * np + col0 + lr] = acc[v];
}

// ---------------- generic fp64 GEMM (one thread / element) ----------------
__global__ void dgemm_k(const double* __restrict__ A, const double* __restrict__ B,
                        double* __restrict__ C, int m, int n, int k, int ta, int tb) {
  int idx = blockIdx.x * blockDim.x + threadIdx.x;
  if (idx >= m * n) return;
  int i = idx / n, j = idx % n;
  double s = 0.0;
  for (int p = 0; p < k; ++p) {
    double av = ta ? A[p * m + i] : A[i * k + p];
    double bv = tb ? B[j * k + p] : B[p * n + j];
    s += av * bv;
  }
  C[idx] = s;
}

__global__ void sym_k(const double* __restrict__ T, double* __restrict__ C) {
  int idx = blockIdx.x * blockDim.x + threadIdx.x;
  if (idx >= Nn * Nn) return;
  int i = idx / Nn, j = idx % Nn;
  C[idx] = 0.5 * (T[i * Nn + j] + T[j * Nn + i]);
}

// ---------------- svec / smat ----------------
__global__ void smat_k(const double* __restrict__ v, double* __restrict__ Mo) {
  int idx = blockIdx.x * blockDim.x + threadIdx.x;
  if (idx >= Nn * Nn) return;
  int i = idx / Nn, j = idx % Nn;
  int r = i > j ? i : j, cc = i > j ? j : i;
  double val = v[r * (r + 1) / 2 + cc];
  if (i != j) val *= DSQH;
  Mo[idx] = val;
}

__global__ void svec_sym_k(const double* __restrict__ T, double* __restrict__ vout) {
  int k = blockIdx.x * blockDim.x + threadIdx.x;
  if (k >= NT) return;
  int r, cc; tril_rc(k, r, cc);
  double val = 0.5 * (T[r * Nn + cc] + T[cc * Nn + r]);
  if (r != cc) val *= DSQ2;
  vout[k] = val;
}

// ---------------- ipm1 ----------------
__global__ void build_A_k(const float* __restrict__ A1, double* __restrict__ Ad) {
  int idx = blockIdx.x * blockDim.x + threadIdx.x;
  if (idx >= Mm * NT) return;
  int m = idx / NT, k = idx % NT;
  int r, cc; tril_rc(k, r, cc);
  double w = (r == cc) ? 1.0 : DSQ2;
  Ad[idx] = w * (double)A1[(size_t)m * Nn * Nn + r * Nn + cc];
}

__global__ void build_cvec_k(const float* __restrict__ cIn, double* __restrict__ cvec) {
  int k = blockIdx.x * blockDim.x + threadIdx.x;
  if (k >= NT) return;
  int r, cc; tril_rc(k, r, cc);
  double w = (r == cc) ? 1.0 : DSQ2;
  cvec[k] = w * (double)cIn[r * Nn + cc];
}

__global__ void rownorm_k(const double* __restrict__ Ad, double* __restrict__ rown) {
  int m = blockIdx.x;
  double s = 0.0;
  for (int k = threadIdx.x; k < NT; k += 256) { double v = Ad[(size_t)m * NT + k]; s += v * v; }
  s = blockSum256(s);
  if (threadIdx.x == 0) rown[m] = sqrt(s);
}

__global__ void ipm1_scalars_k(const double* __restrict__ Ad, const double* __restrict__ cvec,
                               const float* __restrict__ bIn, const double* __restrict__ rown,
                               double* __restrict__ scal) {
  int t = threadIdx.x;
  double mx = -1e300;
  for (int m = t; m < Mm; m += 256)
    mx = fmax(mx, (1.0 + fabs((double)bIn[m])) / (1.0 + rown[m]));
  double ksi = 70.0 * blockMax256(mx);
  mx = -1e300;
  for (int m = t; m < Mm; m += 256) mx = fmax(mx, rown[m]);
  double maxrown = blockMax256(mx);
  double s = 0.0;
  for (int k = t; k < NT; k += 256) s += cvec[k] * cvec[k];
  double cn = sqrt(blockSum256(s));
  s = 0.0;
  for (long i = t; i < (long)Mm * NT; i += 256) s += Ad[i] * Ad[i];
  double fA = sqrt(blockSum256(s));
  s = 0.0;
  for (int m = t; m < Mm; m += 256) s += (double)bIn[m] * (double)bIn[m];
  double bn = sqrt(blockSum256(s));
  if (t == 0) {
    scal[S_KSI]    = ksi;
    scal[S_ETA]    = (1.0 + fmax(maxrown, cn)) / sqrt(70.0);
    scal[S_NORMSA] = fmax(1.0, sqrt(fA));
    scal[S_NORMB]  = fmax(1.0, bn);
    scal[S_NORMC]  = fmax(1.0, cn);
  }
}

__global__ void normalizeA_k(double* __restrict__ Ad, const double* __restrict__ scal) {
  int idx = blockIdx.x * blockDim.x + threadIdx.x;
  if (idx >= Mm * NT) return;
  Ad[idx] /= scal[S_NORMSA];
}

__global__ void init_vecs_k(const float* __restrict__ bIn, double* __restrict__ cvec,
                            double* __restrict__ x, double* __restrict__ z,
                            double* __restrict__ b2, const double* __restrict__ scal) {
  int idx = blockIdx.x * blockDim.x + threadIdx.x;
  if (idx >= NT) return;
  double normsA = scal[S_NORMSA], normc = scal[S_NORMC];
  int r, cc; tril_rc(idx, r, cc);
  cvec[idx] = cvec[idx] / (normc * normsA);
  double diag = (r == cc) ? 1.0 : 0.0;
  x[idx] = diag * scal[S_KSI] * normsA;
  z[idx] = diag * scal[S_ETA] / (normc * normsA);
  if (idx < Mm) b2[idx] = (double)bIn[idx] / scal[S_NORMB];
}

// ---------------- factorizations / solves ----------------
__global__ void chol_upper_k(const double* __restrict__ X, double* __restrict__ U) {
  int t = threadIdx.x; const int nt = 256; const int n = Nn;
  for (int idx = t; idx < n * n; idx += nt) U[idx] = X[idx];
  __syncthreads();
  for (int k = 0; k < n; ++k) {
    if (t == 0) U[k * n + k] = sqrt(U[k * n + k]);
    __syncthreads();
    double dk = U[k * n + k];
    for (int j = k + 1 + t; j < n; j += nt) U[k * n + j] /= dk;
    __syncthreads();
    int w = n - k - 1;
    for (int idx = t; idx < w * w; idx += nt) {
      int i = k + 1 + idx / w, j = k + 1 + idx % w;
      if (j >= i) U[i * n + j] -= U[k * n + i] * U[k * n + j];
    }
    __syncthreads();
  }
  for (int idx = t; idx < n * n; idx += nt) {
    int i = idx / n, j = idx % n;
    if (i > j) U[idx] = 0.0;
  }
}

__global__ void triinv_upper_k(const double* __restrict__ U, double* __restrict__ Ri) {
  int j = blockIdx.x * blockDim.x + threadIdx.x;
  const int n = Nn;
  if (j >= n) return;
  for (int i = 0; i < n; ++i) Ri[i * n + j] = 0.0;
  Ri[j * n + j] = 1.0 / U[j * n + j];
  for (int i = j - 1; i >= 0; --i) {
    double s = 0.0;
    for (int p = i + 1; p <= j; ++p) s += U[i * n + p] * Ri[p * n + j];
    Ri[i * n + j] = -s / U[i * n + i];
  }
}

__global__ void solve_k(const double* __restrict__ Min, const double* __restrict__ hIn,
                        double* __restrict__ dyOut, double* __restrict__ Mw,
                        double* __restrict__ hw) {
  int t = threadIdx.x; const int nt = 256; const int n = Mm;
  for (int idx = t; idx < n * n; idx += nt) Mw[idx] = Min[idx];
  for (int i = t; i < n; i += nt) hw[i] = hIn[i];
  __syncthreads();
  for (int k = 0; k < n; ++k) {
    double pk = Mw[k * n + k];
    for (int i = k + 1 + t; i < n; i += nt) {
      double f = Mw[i * n + k] / pk;
      for (int j = k; j < n; ++j) Mw[i * n + j] -= f * Mw[k * n + j];
      hw[i] -= f * hw[k];
    }
    __syncthreads();
  }
  if (t == 0) {
    for (int i = n - 1; i >= 0; --i) {
      double s = hw[i];
      for (int j = i + 1; j < n; ++j) s -= Mw[i * n + j] * dyOut[j];
      dyOut[i] = s / Mw[i * n + i];
    }
  }
}

// smallest eigenvalue via Gershgorin-shifted power iteration (deterministic, branch-free)
__global__ void mineig_k(const double* __restrict__ W, double* __restrict__ scal, int outIdx) {
  __shared__ double v[Nn], wv[Nn], red[256];
  int t = threadIdx.x; const int n = Nn;
  double local = -1e300;
  for (int i = t; i < n; i += 256) {
    double s = 0.0;
    for (int j = 0; j < n; ++j) s += (j == i) ? W[i * n + j] : fabs(W[i * n + j]);
    local = fmax(local, s);
  }
  red[t] = local; __syncthreads();
  for (int s2 = 128; s2 > 0; s2 >>= 1) { if (t < s2) red[t] = fmax(red[t], red[t + s2]); __syncthreads(); }
  double sh = red[0]; __syncthreads();
  for (int i = t; i < n; i += 256) v[i] = 1.0 / (1.0 + i);
  __syncthreads();
  for (int it = 0; it < 200; ++it) {
    for (int i = t; i < n; i += 256) {
      double s = 0.0;
      for (int j = 0; j < n; ++j) s += W[i * n + j] * v[j];
      wv[i] = sh * v[i] - s;           // B = sh*I - W  (PSD)
    }
    __syncthreads();
    double nl = 0.0;
    for (int i = t; i < n; i += 256) nl += wv[i] * wv[i];
    red[t] = nl; __syncthreads();
    for (int s2 = 128; s2 > 0; s2 >>= 1) { if (t < s2) red[t] += red[t + s2]; __syncthreads(); }
    double inv = (red[0] > 0.0) ? 1.0 / sqrt(red[0]) : 0.0; __syncthreads();
    for (int i = t; i < n; i += 256) v[i] = wv[i] * inv;
    __syncthreads();
  }
  double acc = 0.0;
  for (int i = t; i < n; i += 256) {
    double s = 0.0;
    for (int j = 0; j < n; ++j) s += W[i * n + j] * v[j];
    acc += v[i] * (sh * v[i] - s);
  }
  red[t] = acc; __syncthreads();
  for (int s2 = 128; s2 > 0; s2 >>= 1) { if (t < s2) red[t] += red[t + s2]; __syncthreads(); }
  if (t == 0) scal[outIdx] = sh - red[0];   // lambda_min(W)
}

__global__ void step_final_k(double* __restrict__ scal, int lamIdx, int outIdx,
                             double gconst, int gIdx) {
  double lam = scal[lamIdx];
  double g = (gIdx >= 0) ? scal[gIdx] : gconst;
  double s = (lam < 0.0) ? (-1.0 / lam) : 20.0;
  scal[outIdx] = fmin(0.5, g * s);
}

// ---------------- residuals / reductions ----------------
__global__ void rp_k(const double* __restrict__ Ad, const double* __restrict__ x,
                     const double* __restrict__ b2, double* __restrict__ rp) {
  int m = blockIdx.x;
  double s = 0.0;
  const double* row = Ad + (size_t)m * NT;
  for (int k = threadIdx.x; k < NT; k += 256) s += row[k] * x[k];
  s = blockSum256(s);
  if (threadIdx.x == 0) rp[m] = b2[m] - s;
}

__global__ void Rd0_k(const double* __restrict__ cvec, const double* __restrict__ z,
                      double* __restrict__ Rd) {
  int k = blockIdx.x * blockDim.x + threadIdx.x;
  if (k >= NT) return;
  Rd[k] = cvec[k] - z[k];                 // y == 0 at this point
}

__global__ void mu_k(const double* __restrict__ x, const double* __restrict__ z,
                     double* __restrict__ scal) {
  double s = 0.0;
  for (int k = threadIdx.x; k < NT; k += 256) s += x[k] * z[k];
  s = blockSum256(s);
  if (threadIdx.x == 0) { scal[S_XZ] = s; scal[S_MU] = s / 70.0; }
}

__global__ void h_k(const double* __restrict__ Ad, const double* __restrict__ rp,
                    const double* __restrict__ va, const double* __restrict__ vb,
                    double* __restrict__ h) {
  int m = blockIdx.x;
  double s = 0.0;
  const double* row = Ad + (size_t)m * NT;
  for (int k = threadIdx.x; k < NT; k += 256) s += row[k] * (va[k] - vb[k]);
  s = blockSum256(s);
  if (threadIdx.x == 0) h[m] = rp[m] + s;
}

__global__ void dz_k(const double* __restrict__ Ad, const double* __restrict__ Rd,
                     const double* __restrict__ dy, double* __restrict__ dzOut) {
  int k = blockIdx.x * blockDim.x + threadIdx.x;
  if (k >= NT) return;
  double s = 0.0;
  for (int m = 0; m < Mm; ++m) s += Ad[(size_t)m * NT + k] * dy[m];
  dzOut[k] = Rd[k] - s;
}

__global__ void dx_pred_k(const double* __restrict__ x, const double* __restrict__ skdz,
                          double* __restrict__ dx) {
  int k = blockIdx.x * blockDim.x + threadIdx.x;
  if (k >= NT) return;
  dx[k] = -x[k] - skdz[k];
}

__global__ void dx_corr_k(const double* __restrict__ skRc2, const double* __restrict__ skdz2,
                          double* __restrict__ dx2) {
  int k = blockIdx.x * blockDim.x + threadIdx.x;
  if (k >= NT) return;
  dx2[k] = skRc2[k] - skdz2[k];
}

__global__ void rc_pred_k(const double* __restrict__ hpXZ, double* __restrict__ Rc) {
  int k = blockIdx.x * blockDim.x + threadIdx.x;
  if (k >= NT) return;
  int r, cc; tril_rc(k, r, cc);
  double w = (r == cc) ? 1.0 : DSQ2;
  Rc[k] = -w * hpXZ[r * Nn + cc];
}

__global__ void rc_corr_k(const double* __restrict__ hpXZ, const double* __restrict__ hpDD,
                          const double* __restrict__ scal, double* __restrict__ Rc2) {
  int k = blockIdx.x * blockDim.x + threadIdx.x;
  if (k >= NT) return;
  int r, cc; tril_rc(k, r, cc);
  double base = -(hpXZ[r * Nn + cc] + hpDD[r * Nn + cc]);
  if (r == cc) base += scal[S_SIGMA] * scal[S_MU];
  double w = (r == cc) ? 1.0 : DSQ2;
  Rc2[k] = w * base;
}

__global__ void sigma_k(const double* __restrict__ x, const double* __restrict__ z,
                        const double* __restrict__ dx, const double* __restrict__ dz,
                        double* __restrict__ scal) {
  int t = threadIdx.x;
  double a = scal[S_ALPHA], b = scal[S_BETA];
  double s = 0.0;
  for (int k = t; k < NT; k += 256) s += (x[k] + a * dx[k]) * (z[k] + b * dz[k]);
  s = blockSum256(s);
  if (t == 0) {
    double inner = s;
    double mab = fmin(a, b);
    double mu = scal[S_MU], xz = scal[S_XZ];
    double e = 1.0;
    if (mu > 1e-6) e = (mab < sqrt(1.0 / 3.0)) ? 1.0 : fmax(1.0, 3.0 * mab * mab);
    double frac = inner / xz;
    double sig = (inner < 0.0) ? 0.8 : fmin(1.0, pow(fmax(frac, 1e-30), e));
    scal[S_INNER] = inner;
    scal[S_SIGMA] = sig;
    scal[S_MAB]   = mab;
    scal[S_G2]    = 0.9 + 0.09 * mab;
  }
}

// ---------------- fp32 padded fills for the WMMA path ----------------
__global__ void fill_Sb_k(const double* __restrict__ Ad, float* __restrict__ Sb) {
  int idx = blockIdx.x * blockDim.x + threadIdx.x;
  if (idx >= Mm * NP * NP) return;
  int l = idx / (NP * NP), rem = idx % (NP * NP);
  int i = rem / NP, j = rem % NP;
  float v = 0.0f;
  if (i < Nn && j < Nn) {
    int r = i > j ? i : j, cc = i > j ? j : i;
    double val = Ad[(size_t)l * NT + r * (r + 1) / 2 + cc];
    if (i != j) val *= DSQH;
    v = (float)val;
  }
  Sb[idx] = v;
}

__global__ void fill_pad80_k(const double* __restrict__ src, float* __restrict__ dst) {
  int idx = blockIdx.x * blockDim.x + threadIdx.x;
  if (idx >= NP * NP) return;
  int i = idx / NP, j = idx % NP;
  dst[idx] = (i < Nn && j < Nn) ? (float)src[i * Nn + j] : 0.0f;
}

__global__ void fill_Af_k(const double* __restrict__ Ad, float* __restrict__ Af) {
  int idx = blockIdx.x * blockDim.x + threadIdx.x;
  if (idx >= MP * NTP) return;
  int m = idx / NTP, k = idx % NTP;
  Af[idx] = (m < Mm && k < NT) ? (float)Ad[(size_t)m * NT + k] : 0.0f;
}

// Skf[k][l] = svec(sym(T2b[l]))[k]  (NTP x MP, zero padded)
__global__ void fill_Skf_k(const float* __restrict__ T2b, float* __restrict__ Skf) {
  int idx = blockIdx.x * blockDim.x + threadIdx.x;
  if (idx >= NTP * MP) return;
  int k = idx / MP, l = idx % MP;
  float v = 0.0f;
  if (k < NT && l < Mm) {
    int r, cc; tril_rc(k, r, cc);
    const float* T = T2b + (size_t)l * NP * NP;
    float val = 0.5f * (T[r * NP + cc] + T[cc * NP + r]);
    if (r != cc) val *= (float)DSQ2;
    v = val;
  }
  Skf[idx] = v;
}

__global__ void Mf2Mmat_k(const float* __restrict__ Mf, double* __restrict__ Mmat) {
  int idx = blockIdx.x * blockDim.x + threadIdx.x;
  if (idx >= Mm * Mm) return;
  int i = idx / Mm, j = idx % Mm;
  Mmat[idx] = (double)Mf[i * MP + j];
}

// ---------------- final update + output (float, per harness output dtype) ----------------
__global__ void out_k(const double* __restrict__ x, const double* __restrict__ dx2,
                      const double* __restrict__ z, const double* __restrict__ dz2,
                      const double* __restrict__ dy2, const double* __restrict__ scal,
                      float* __restrict__ out) {
  int idx = blockIdx.x * blockDim.x + threadIdx.x;
  if (idx >= 2 * NT + Mm) return;
  double a = scal[S_ALPHA], b = scal[S_BETA], nc = scal[S_NORMC];
  if (idx < NT)           out[idx] = (float)(x[idx] + a * dx2[idx]);
  else if (idx < 2 * NT)  { int k = idx - NT; out[idx] = (float)(z[k] + b * dz2[k]); }
  else                    { int m = idx - 2 * NT; out[idx] = (float)(b * dy2[m] * nc); }
}

// ---------------- host driver ----------------
extern "C" void kernel_launch(void* const* d_in, const int* in_sizes, int n_in,
                              void* d_out, int out_size, void* d_ws, size_t ws_size,
                              hipStream_t stream) {
  (void)in_sizes; (void)n_in; (void)out_size; (void)ws_size;
  const float* A1 = (const float*)d_in[0];   // (M,N,N)
  const float* bI = (const float*)d_in[1];   // (M,1)
  const float* cI = (const float*)d_in[2];   // (N,N)
  float* out = (float*)d_out;

  // ---- bump allocator over d_ws ----
  char* p = (char*)d_ws;
  auto allocd = [&](size_t n) { double* r = (double*)p; p += n * sizeof(double); return r; };
  auto allocf = [&](size_t n) { float* r = (float*)p; p += n * sizeof(float); return r; };

  double* Ad   = allocd((size_t)Mm * NT);
  double* cvec = allocd(NT);
  double* b2   = allocd(Mm);
  double* x    = allocd(NT);
  double* z    = allocd(NT);
  double* rown = allocd(Mm);
  double* scal = allocd(64);
  double* Xm   = allocd(Nn * Nn); double* Zm   = allocd(Nn * Nn);
  double* Qm   = allocd(Nn * Nn); double* Pm   = allocd(Nn * Nn);
  double* Qinv = allocd(Nn * Nn); double* Pinv = allocd(Nn * Nn);
  double* Zinv = allocd(Nn * Nn); double* XZ   = allocd(Nn * Nn);
  double* Tt   = allocd(Nn * Nn); double* T2t  = allocd(Nn * Nn);
  double* hpXZ = allocd(Nn * Nn); double* hpDD = allocd(Nn * Nn);
  double* Dsm  = allocd(Nn * Nn); double* dxm  = allocd(Nn * Nn);
  double* dzm  = allocd(Nn * Nn); double* DD   = allocd(Nn * Nn);
  double* W1   = allocd(Nn * Nn); double* W2   = allocd(Nn * Nn);
  double* rp   = allocd(Mm);
  double* Rd   = allocd(NT); double* Rc  = allocd(NT); double* Rc2 = allocd(NT);
  double* hv   = allocd(Mm);
  double* dy   = allocd(Mm); double* dy2 = allocd(Mm);
  double* dzv  = allocd(NT); double* dz2 = allocd(NT);
  double* dxv  = allocd(NT); double* dx2 = allocd(NT);
  double* skRd = allocd(NT); double* skRc = allocd(NT); double* skdz = allocd(NT);
  double* skRc2 = allocd(NT); double* skdz2 = allocd(NT);
  double* Mmat = allocd(Mm * Mm); double* Mw = allocd(Mm * Mm); double* hw = allocd(Mm);
  float* Sb  = allocf((size_t)Mm * NP * NP);
  float* T1b = allocf((size_t)Mm * NP * NP);
  float* T2b = allocf((size_t)Mm * NP * NP);
  float* Xf  = allocf(NP * NP); float* Zif = allocf(NP * NP);
  float* Af  = allocf((size_t)MP * NTP);
  float* Skf = allocf((size_t)NTP * MP);
  float* Mf  = allocf(MP * MP);

  const int TB = 256;
  auto cd = [](int a, int b) { return (a + b - 1) / b; };
  const int gNN = cd(Nn * Nn, TB), gNT = cd(NT, TB);

  auto dgemm = [&](const double* A, const double* B, double* C, int m, int n, int k,
                   int ta, int tb) {
    dgemm_k<<<cd(m * n, TB), TB, 0, stream>>>(A, B, C, m, n, k, ta, tb);
  };
  // skmult with single column: vout = svec(sym(K @ smat(vin) @ op(G)))
  auto skmultL1 = [&](const double* K, const double* G, int tbG, const double* vin,
                      double* vout) {
    smat_k<<<gNN, TB, 0, stream>>>(vin, Dsm);
    dgemm(K, Dsm, Tt, Nn, Nn, Nn, 0, 0);
    dgemm(Tt, G, T2t, Nn, Nn, Nn, 0, tbG);
    svec_sym_k<<<gNT, TB, 0, stream>>>(T2t, vout);
  };
  // hp(U,P) = sym(P @ U @ Pinv)
  auto hpOf = [&](const double* U, double* outM) {
    dgemm(Pm, U, Tt, Nn, Nn, Nn, 0, 0);
    dgemm(Tt, Pinv, T2t, Nn, Nn, Nn, 0, 0);
    sym_k<<<gNN, TB, 0, stream>>>(T2t, outM);
  };
  // step_scale: lambda_min(Rinv^T smat(dv) Rinv) -> alpha/beta slot
  auto stepScale = [&](const double* Rinv, const double* dv, double gconst, int gIdx,
                       int outIdx) {
    smat_k<<<gNN, TB, 0, stream>>>(dv, Dsm);
    dgemm(Rinv, Dsm, W1, Nn, Nn, Nn, 1, 0);
    dgemm(W1, Rinv, W2, Nn, Nn, Nn, 0, 0);
    mineig_k<<<1, 256, 0, stream>>>(W2, scal, S_LAM);
    step_final_k<<<1, 1, 0, stream>>>(scal, S_LAM, outIdx, gconst, gIdx);
  };

  // ================= ipm1 =================
  build_A_k<<<cd(Mm * NT, TB), TB, 0, stream>>>(A1, Ad);
  build_cvec_k<<<gNT, TB, 0, stream>>>(cI, cvec);
  rownorm_k<<<Mm, 256, 0, stream>>>(Ad, rown);
  ipm1_scalars_k<<<1, 256, 0, stream>>>(Ad, cvec, bI, rown, scal);
  normalizeA_k<<<cd(Mm * NT, TB), TB, 0, stream>>>(Ad, scal);
  init_vecs_k<<<gNT, TB, 0, stream>>>(bI, cvec, x, z, b2, scal);

  // ================= ipm2: factorizations =================
  smat_k<<<gNN, TB, 0, stream>>>(x, Xm);
  smat_k<<<gNN, TB, 0, stream>>>(z, Zm);
  chol_upper_k<<<1, 256, 0, stream>>>(Xm, Qm);
  chol_upper_k<<<1, 256, 0, stream>>>(Zm, Pm);
  triinv_upper_k<<<1, 128, 0, stream>>>(Qm, Qinv);
  triinv_upper_k<<<1, 128, 0, stream>>>(Pm, Pinv);
  dgemm(Pinv, Pinv, Zinv, Nn, Nn, Nn, 0, 1);        // Zinv = Pinv @ Pinv^T
  rp_k<<<Mm, 256, 0, stream>>>(Ad, x, b2, rp);
  Rd0_k<<<gNT, TB, 0, stream>>>(cvec, z, Rd);
  mu_k<<<1, 256, 0, stream>>>(x, z, scal);
  dgemm(Xm, Zm, XZ, Nn, Nn, Nn, 0, 0);
  hpOf(XZ, hpXZ);
  rc_pred_k<<<gNT, TB, 0, stream>>>(hpXZ, Rc);

  // ================= Schur complement via WMMA =================
  fill_Sb_k<<<cd(Mm * NP * NP, TB), TB, 0, stream>>>(Ad, Sb);
  fill_pad80_k<<<cd(NP * NP, TB), TB, 0, stream>>>(Xm, Xf);
  fill_pad80_k<<<cd(NP * NP, TB), TB, 0, stream>>>(Zinv, Zif);
  {
    dim3 g(NP / 16, NP / 16, Mm);
    wmma_bgemm_k<<<g, 32, 0, stream>>>(Zif, Sb, T1b, NP, NP, NP,
                                       0L, (long)NP * NP, (long)NP * NP);
    wmma_bgemm_k<<<g, 32, 0, stream>>>(T1b, Xf, T2b, NP, NP, NP,
                                       (long)NP * NP, 0L, (long)NP * NP);
  }
  fill_Skf_k<<<cd(NTP * MP, TB), TB, 0, stream>>>(T2b, Skf);
  fill_Af_k<<<cd(MP * NTP, TB), TB, 0, stream>>>(Ad, Af);
  {
    dim3 g(MP / 16, MP / 16, 1);
    wmma_bgemm_k<<<g, 32, 0, stream>>>(Af, Skf, Mf, MP, NTP, MP, 0L, 0L, 0L);
  }
  Mf2Mmat_k<<<cd(Mm * Mm, TB), TB, 0, stream>>>(Mf, Mmat);

  // ================= predictor =================
  skmultL1(Zinv, Xm, 0, Rd, skRd);                  // skmult(X, Zinv, Rd)
  skmultL1(Pinv, Pinv, 1, Rc, skRc);                // skmult(Pinv, Pinv, Rc)
  h_k<<<Mm, 256, 0, stream>>>(Ad, rp, skRd, skRc, hv);
  solve_k<<<1, 256, 0, stream>>>(Mmat, hv, dy, Mw, hw);
  dz_k<<<gNT, TB, 0, stream>>>(Ad, Rd, dy, dzv);
  skmultL1(Zinv, Xm, 0, dzv, skdz);
  dx_pred_k<<<gNT, TB, 0, stream>>>(x, skdz, dxv);
  stepScale(Qinv, dxv, 0.9, -1, S_ALPHA);
  stepScale(Pinv, dzv, 0.9, -1, S_BETA);
  sigma_k<<<1, 256, 0, stream>>>(x, z, dxv, dzv, scal);

  // ================= corrector =================
  smat_k<<<gNN, TB, 0, stream>>>(dxv, dxm);
  smat_k<<<gNN, TB, 0, stream>>>(dzv, dzm);
  dgemm(dxm, dzm, DD, Nn, Nn, Nn, 0, 0);
  hpOf(DD, hpDD);
  rc_corr_k<<<gNT, TB, 0, stream>>>(hpXZ, hpDD, scal, Rc2);
  skmultL1(Pinv, Pinv, 1, Rc2, skRc2);
  h_k<<<Mm, 256, 0, stream>>>(Ad, rp, skRd, skRc2, hv);
  solve_k<<<1, 256, 0, stream>>>(Mmat, hv, dy2, Mw, hw);
  dz_k<<<gNT, TB, 0, stream>>>(Ad, Rd, dy2, dz2);
  skmultL1(Zinv, Xm, 0, dz2, skdz2);
  dx_corr_k<<<gNT, TB, 0, stream>>>(skRc2, skdz2, dx2);
  stepScale(Qinv, dx2, 0.0, S_G2, S_ALPHA);
  stepScale(Pinv, dz2, 0.0, S_G2, S_BETA);

  // ================= output =================
  out_k<<<cd(2 * NT + Mm, TB), TB, 0, stream>>>(x, dx2, z, dz2, dy2, scal, out);
}